// Encoder_59777354825729
// MI455X (gfx1250) — compile-verified
//
#include <hip/hip_runtime.h>
#include <hip/hip_bf16.h>
#include <math.h>

// ---------------------------------------------------------------------------
// 2-layer BiLSTM, B=256, S=2048, F=64, H1=4 (16 gates), H2=2 (8 gates).
// Stage 1: WMMA f32 16x16x4 projection px = x @ Wih^T + bih + bhh (both dirs,
//          one read of x). Stage 2/3: per-sequence serial scans, one thread
//          per (batch, direction), coalesced px[s][b][g] layout.
// ---------------------------------------------------------------------------

#define BATCH 256
#define SEQ   2048
#define FIN   64

typedef float v2f __attribute__((ext_vector_type(2)));
typedef float v8f __attribute__((ext_vector_type(8)));

__device__ __forceinline__ float sigmoid_fast(float x) {
    return 1.0f / (1.0f + __expf(-x));
}
__device__ __forceinline__ float tanh_fast(float x) {
    // robust at +/-inf: exp(2x)=inf -> 1 ; exp(2x)=0 -> -1
    return 1.0f - 2.0f / (__expf(2.0f * x) + 1.0f);
}

// ---------------------------------------------------------------------------
// Stage 1: px[dir][s*256+b][g] = sum_f x[b,s,f]*Wih[g,f] + bih[g] + bhh[g]
// One wave per 16-row tile of the flattened [B*S, 64] matrix.
// 16 chained V_WMMA_F32_16X16X4_F32 per direction, bias preloaded into C.
// ---------------------------------------------------------------------------
__global__ __launch_bounds__(256) void proj1_wmma(
    const float* __restrict__ x,
    const float* __restrict__ WihF, const float* __restrict__ WihB,
    const float* __restrict__ bihF, const float* __restrict__ bhhF,
    const float* __restrict__ bihB, const float* __restrict__ bhhB,
    float* __restrict__ pxF, float* __restrict__ pxB)
{
    const int wave    = (blockIdx.x * blockDim.x + threadIdx.x) >> 5; // tile id
    const int lane    = threadIdx.x & 31;
    const int m       = lane & 15;   // M row (A) / N col (B,C,D)
    const int hi      = lane >> 4;   // lane half selects K pair
    const int rowBase = wave * 16;

    // B fragments: lane holds Wih[n=m][4*kk + 2*hi + {0,1}] for each K-step kk
    v2f bf[16], bb[16];
#pragma unroll
    for (int kk = 0; kk < 16; ++kk) {
        const int col = 4 * kk + 2 * hi;
        bf[kk] = *(const v2f*)(WihF + m * FIN + col);
        bb[kk] = *(const v2f*)(WihB + m * FIN + col);
    }

    const float biasF = bihF[m] + bhhF[m];
    const float biasB = bihB[m] + bhhB[m];
    v8f accF, accB;
#pragma unroll
    for (int v = 0; v < 8; ++v) { accF[v] = biasF; accB[v] = biasB; }

    // A fragments straight from global: lane reads row (rowBase+m),
    // cols 4*kk + 2*hi + {0,1}
    const float* rowPtr = x + (size_t)(rowBase + m) * FIN;
#pragma unroll
    for (int kk = 0; kk < 16; ++kk) {
        const int col = 4 * kk + 2 * hi;
        v2f a = *(const v2f*)(rowPtr + col);
        accF = __builtin_amdgcn_wmma_f32_16x16x4_f32(
            false, a, false, bf[kk], (short)0, accF, false, false);
        accB = __builtin_amdgcn_wmma_f32_16x16x4_f32(
            false, a, false, bb[kk], (short)0, accB, false, false);
    }

    // Scatter D to px[s*256 + b][g]; row r = b*SEQ + s (x is [B,S,F])
#pragma unroll
    for (int v = 0; v < 8; ++v) {
        const int r = rowBase + v + 8 * hi;
        const int b = r >> 11;       // r / 2048
        const int s = r & 2047;      // r % 2048
        const size_t o = ((size_t)(s * BATCH + b) << 4) + m;
        pxF[o] = accF[v];
        pxB[o] = accB[v];
    }
}

// ---------------------------------------------------------------------------
// Stage 2: layer-1 recurrence. 512 threads: t = dir*256 + b.
// Gate order (i,f,g,o), H1=4. Writes h1[(s*256+b)*8 + dir*4 + j].
// ---------------------------------------------------------------------------
__global__ __launch_bounds__(256) void scan1(
    const float* __restrict__ pxF, const float* __restrict__ pxB,
    const float* __restrict__ WhhF, const float* __restrict__ WhhB,
    float* __restrict__ h1)
{
    const int t   = blockIdx.x * blockDim.x + threadIdx.x;
    const int b   = t & 255;
    const int dir = t >> 8;

    const float* px  = dir ? pxB : pxF;
    const float* Whh = dir ? WhhB : WhhF;   // [16][4], wave-uniform -> SGPRs

    float w[64];
#pragma unroll
    for (int i = 0; i < 64; ++i) w[i] = Whh[i];

    float h[4] = {0.f, 0.f, 0.f, 0.f};
    float c[4] = {0.f, 0.f, 0.f, 0.f};

    for (int step = 0; step < SEQ; ++step) {
        const int s = dir ? (SEQ - 1 - step) : step;
        const float4* p = (const float4*)(px + ((size_t)(s * BATCH + b) << 4));
        float4 g0 = p[0], g1 = p[1], g2 = p[2], g3 = p[3];
        float gates[16] = {g0.x, g0.y, g0.z, g0.w,
                           g1.x, g1.y, g1.z, g1.w,
                           g2.x, g2.y, g2.z, g2.w,
                           g3.x, g3.y, g3.z, g3.w};
#pragma unroll
        for (int g = 0; g < 16; ++g) {
            gates[g] += w[g * 4 + 0] * h[0] + w[g * 4 + 1] * h[1] +
                        w[g * 4 + 2] * h[2] + w[g * 4 + 3] * h[3];
        }
#pragma unroll
        for (int j = 0; j < 4; ++j) {
            const float ig = sigmoid_fast(gates[j]);
            const float fg = sigmoid_fast(gates[4 + j]);
            const float gg = tanh_fast(gates[8 + j]);
            const float og = sigmoid_fast(gates[12 + j]);
            c[j] = fg * c[j] + ig * gg;
            h[j] = og * tanh_fast(c[j]);
        }
        float4 hv = {h[0], h[1], h[2], h[3]};
        *(float4*)(h1 + (size_t)(s * BATCH + b) * 8 + dir * 4) = hv;
    }
}

// ---------------------------------------------------------------------------
// Stage 3: layer-2 BiLSTM fused (projection + recurrence), input dim 8, H2=2.
// 512 threads: t = dir*256 + b. Output [B, S, 4] = concat(fwd2, bwd2).
// ---------------------------------------------------------------------------
__global__ __launch_bounds__(256) void scan2(
    const float* __restrict__ h1,
    const float* __restrict__ WihF, const float* __restrict__ WhhF,
    const float* __restrict__ bihF, const float* __restrict__ bhhF,
    const float* __restrict__ WihB, const float* __restrict__ WhhB,
    const float* __restrict__ bihB, const float* __restrict__ bhhB,
    float* __restrict__ out)
{
    const int t   = blockIdx.x * blockDim.x + threadIdx.x;
    const int b   = t & 255;
    const int dir = t >> 8;

    const float* Wih = dir ? WihB : WihF;   // [8][8]
    const float* Whh = dir ? WhhB : WhhF;   // [8][2]
    const float* bih = dir ? bihB : bihF;   // [8]
    const float* bhh = dir ? bhhB : bhhF;   // [8]

    float wih[64], whh[16], bsum[8];
#pragma unroll
    for (int i = 0; i < 64; ++i) wih[i] = Wih[i];
#pragma unroll
    for (int i = 0; i < 16; ++i) whh[i] = Whh[i];
#pragma unroll
    for (int i = 0; i < 8; ++i)  bsum[i] = bih[i] + bhh[i];

    float h[2] = {0.f, 0.f};
    float c[2] = {0.f, 0.f};

    for (int step = 0; step < SEQ; ++step) {
        const int s = dir ? (SEQ - 1 - step) : step;
        const float4* p = (const float4*)(h1 + (size_t)(s * BATCH + b) * 8);
        float4 x0 = p[0], x1 = p[1];
        const float xin[8] = {x0.x, x0.y, x0.z, x0.w, x1.x, x1.y, x1.z, x1.w};

        float gates[8];
#pragma unroll
        for (int g = 0; g < 8; ++g) {
            float acc = bsum[g];
#pragma unroll
            for (int f = 0; f < 8; ++f) acc += wih[g * 8 + f] * xin[f];
            acc += whh[g * 2 + 0] * h[0] + whh[g * 2 + 1] * h[1];
            gates[g] = acc;
        }
#pragma unroll
        for (int j = 0; j < 2; ++j) {
            const float ig = sigmoid_fast(gates[j]);
            const float fg = sigmoid_fast(gates[2 + j]);
            const float gg = tanh_fast(gates[4 + j]);
            const float og = sigmoid_fast(gates[6 + j]);
            c[j] = fg * c[j] + ig * gg;
            h[j] = og * tanh_fast(c[j]);
        }
        float2 hv = {h[0], h[1]};
        *(float2*)(out + ((size_t)b * SEQ + s) * 4 + dir * 2) = hv;
    }
}

// ---------------------------------------------------------------------------
extern "C" void kernel_launch(void* const* d_in, const int* in_sizes, int n_in,
                              void* d_out, int out_size, void* d_ws, size_t ws_size,
                              hipStream_t stream) {
    (void)in_sizes; (void)n_in; (void)out_size; (void)ws_size;

    const float* x        = (const float*)d_in[0];
    const float* l1_WihF  = (const float*)d_in[1];
    const float* l1_WhhF  = (const float*)d_in[2];
    const float* l1_bihF  = (const float*)d_in[3];
    const float* l1_bhhF  = (const float*)d_in[4];
    const float* l1_WihB  = (const float*)d_in[5];
    const float* l1_WhhB  = (const float*)d_in[6];
    const float* l1_bihB  = (const float*)d_in[7];
    const float* l1_bhhB  = (const float*)d_in[8];
    const float* l2_WihF  = (const float*)d_in[9];
    const float* l2_WhhF  = (const float*)d_in[10];
    const float* l2_bihF  = (const float*)d_in[11];
    const float* l2_bhhF  = (const float*)d_in[12];
    const float* l2_WihB  = (const float*)d_in[13];
    const float* l2_WhhB  = (const float*)d_in[14];
    const float* l2_bihB  = (const float*)d_in[15];
    const float* l2_bhhB  = (const float*)d_in[16];

    const size_t pxElems = (size_t)BATCH * SEQ * 16;   // 8,388,608 floats/dir
    float* ws  = (float*)d_ws;
    float* pxF = ws;
    float* pxB = ws + pxElems;
    float* h1  = ws + 2 * pxElems;                     // [S*B][8]

    // Stage 1: 524288 rows / 16 per wave = 32768 waves; 8 waves/block
    proj1_wmma<<<4096, 256, 0, stream>>>(x, l1_WihF, l1_WihB,
                                         l1_bihF, l1_bhhF, l1_bihB, l1_bhhB,
                                         pxF, pxB);
    // Stage 2: 512 sequences (256 batch x 2 dirs)
    scan1<<<2, 256, 0, stream>>>(pxF, pxB, l1_WhhF, l1_WhhB, h1);
    // Stage 3
    scan2<<<2, 256, 0, stream>>>(h1, l2_WihF, l2_WhhF, l2_bihF, l2_bhhF,
                                 l2_WihB, l2_WhhB, l2_bihB, l2_bhhB,
                                 (float*)d_out);
}